// ViGBlock_687194768121
// MI455X (gfx1250) — compile-verified
//
#include <hip/hip_runtime.h>
#include <math.h>

typedef float v2f __attribute__((ext_vector_type(2)));
typedef float v8f __attribute__((ext_vector_type(8)));

#define BB  16
#define NN  1024
#define CC  192
#define KNN 9
#define HH  768
#define MM  (BB * NN)

__device__ __forceinline__ float gelu_exact(float v) {
    // torch nn.GELU default: exact erf form
    return 0.5f * v * (1.0f + erff(v * 0.7071067811865475f));
}

__device__ __forceinline__ v8f wmma_f32(v2f a, v2f b, v8f c) {
    // V_WMMA_F32_16X16X4_F32: D = A(16x4) * B(4x16) + C(16x16), fp32 throughout
    return __builtin_amdgcn_wmma_f32_16x16x4_f32(
        /*neg_a=*/false, a, /*neg_b=*/false, b,
        /*c_mod=*/(short)0, c, /*reuse_a=*/false, /*reuse_b=*/false);
}

// Generic fused GEMM: Y[M x Nd] = act( X[M x Kd] @ W[Nd x Kd]^T + bias (+ resid) )
// W is torch Linear layout (out, in) row-major, so B[k][n] = W[n*Kd + k].
// One wave computes a 32x32 output tile (2x2 register blocking of 16x16 WMMA
// tiles) -> 1 global b64 load per WMMA instead of 2. K stepped by 4; k-loop
// chunked by 32 (all K here are multiples of 32) with cacheline prefetch.
__global__ __launch_bounds__(128)
void gemm_bias_act(const float* __restrict__ X, const float* __restrict__ W,
                   const float* __restrict__ bias, const float* __restrict__ resid,
                   float* __restrict__ Y, int Nd, int Kd, int act)
{
    const int lane = threadIdx.x & 31;
    const int wave = threadIdx.x >> 5;
    const int lo = lane & 15, hi = lane >> 4;

    const int ntiles_n = Nd >> 5;                 // 32-wide output tiles
    const int tile = blockIdx.x * 4 + wave;
    const int mt = tile / ntiles_n;
    const int nt = tile - mt * ntiles_n;
    const int m0 = mt << 5, n0 = nt << 5;

    // A layout (32-bit 16x4): lane holds row M=lo, K = k + 2*hi + {0,1} in vgpr pair
    const float* xr0 = X + (size_t)(m0 + lo) * Kd + 2 * hi;
    const float* xr1 = xr0 + (size_t)16 * Kd;
    // B layout (4x16): lane holds col N=lo, K = k + 2*hi + {0,1}
    const float* wr0 = W + (size_t)(n0 + lo) * Kd + 2 * hi;
    const float* wr1 = wr0 + (size_t)16 * Kd;

    v8f c00 = {}, c01 = {}, c10 = {}, c11 = {};
    for (int k0 = 0; k0 < Kd; k0 += 32) {
        // Pull the next 128B line of each stream toward the WGP while we compute
        __builtin_prefetch(xr0 + k0 + 32, 0, 3);
        __builtin_prefetch(xr1 + k0 + 32, 0, 3);
        __builtin_prefetch(wr0 + k0 + 32, 0, 3);
        __builtin_prefetch(wr1 + k0 + 32, 0, 3);
        #pragma unroll
        for (int kk = 0; kk < 32; kk += 4) {
            v2f a0 = *(const v2f*)(xr0 + k0 + kk);
            v2f a1 = *(const v2f*)(xr1 + k0 + kk);
            v2f b0 = *(const v2f*)(wr0 + k0 + kk);
            v2f b1 = *(const v2f*)(wr1 + k0 + kk);
            c00 = wmma_f32(a0, b0, c00);
            c01 = wmma_f32(a0, b1, c01);
            c10 = wmma_f32(a1, b0, c10);
            c11 = wmma_f32(a1, b1, c11);
        }
    }

    const float bn0 = bias[n0 + lo];
    const float bn1 = bias[n0 + 16 + lo];
    auto emit = [&](int r, int cidx, float val, float bv) {
        val += bv;
        if (resid) val += resid[(size_t)r * Nd + cidx];
        if (act)   val  = gelu_exact(val);
        Y[(size_t)r * Nd + cidx] = val;
    };
    #pragma unroll
    for (int v = 0; v < 8; ++v) {
        const int ra = m0 + v + 8 * hi;     // C/D layout: row = v + 8*hi, col = lo
        const int rb = ra + 16;
        emit(ra, n0 + lo,      c00[v], bn0);
        emit(ra, n0 + 16 + lo, c01[v], bn1);
        emit(rb, n0 + lo,      c10[v], bn0);
        emit(rb, n0 + 16 + lo, c11[v], bn1);
    }
}

// sim = x @ x^T per batch (fp32 WMMA), streaming top-9 per row.
// One wave per (batch, 16-row strip). 16x16 sim tiles bounce through LDS;
// lanes 0..15 keep a 9-deep insertion-sorted (val,idx) list per row.
__global__ __launch_bounds__(32)
void sim_topk(const float* __restrict__ x, int* __restrict__ graph)
{
    const int lane = threadIdx.x & 31;
    const int lo = lane & 15, hi = lane >> 4;
    const int rt = blockIdx.x & 63;       // N/16 = 64 strips
    const int b  = blockIdx.x >> 6;
    const int r0 = rt << 4;
    const float* xb = x + (size_t)b * NN * CC;

    __shared__ float tile[16 * 16];

    float best[KNN];
    int   bidx[KNN];
    #pragma unroll
    for (int k = 0; k < KNN; ++k) { best[k] = -INFINITY; bidx[k] = 0; }

    const float* arow = xb + (size_t)(r0 + lo) * CC + 2 * hi;
    for (int m0 = 0; m0 < NN; m0 += 16) {
        const float* brow = xb + (size_t)(m0 + lo) * CC + 2 * hi;
        v8f acc = {};
        for (int k0 = 0; k0 < CC; k0 += 4) {
            v2f a  = *(const v2f*)(arow + k0);
            v2f bv = *(const v2f*)(brow + k0);
            acc = wmma_f32(a, bv, acc);
        }
        #pragma unroll
        for (int v = 0; v < 8; ++v)
            tile[(v + 8 * hi) * 16 + lo] = acc[v];
        __syncthreads();

        if (lane < 16) {
            #pragma unroll 1
            for (int j = 0; j < 16; ++j) {
                const float sv = tile[lane * 16 + j];
                if (sv > best[KNN - 1]) {     // strict > : earliest index wins ties
                    int p = KNN - 1;
                    while (p > 0 && sv > best[p - 1]) {
                        best[p] = best[p - 1]; bidx[p] = bidx[p - 1]; --p;
                    }
                    best[p] = sv; bidx[p] = m0 + j;
                }
            }
        }
        __syncthreads();
    }

    if (lane < 16) {
        const int row = b * NN + r0 + lane;
        #pragma unroll
        for (int k = 0; k < KNN; ++k) graph[row * KNN + k] = bidx[k];
    }
}

// agg[b,n,c] = max_k( h[b, graph[b,n,k], c] - h[b,n,c] ),
// then interleave: cat[b,n,2c] = h, cat[b,n,2c+1] = agg   (stack(...,-1).reshape)
__global__ __launch_bounds__(256)
void agg_cat(const float* __restrict__ h, const int* __restrict__ graph,
             float* __restrict__ cat)
{
    const int idx = blockIdx.x * 256 + threadIdx.x;   // over MM*CC (exact multiple)
    const int c  = idx % CC;
    const int bn = idx / CC;
    const int b  = bn / NN;
    const float hv = h[idx];
    float mx = -INFINITY;
    const int* g = graph + (size_t)bn * KNN;
    #pragma unroll
    for (int k = 0; k < KNN; ++k) {
        const int nb = g[k];
        mx = fmaxf(mx, h[((size_t)b * NN + nb) * CC + c] - hv);
    }
    cat[(size_t)bn * (2 * CC) + 2 * c]     = hv;
    cat[(size_t)bn * (2 * CC) + 2 * c + 1] = mx;
}

extern "C" void kernel_launch(void* const* d_in, const int* in_sizes, int n_in,
                              void* d_out, int out_size, void* d_ws, size_t ws_size,
                              hipStream_t stream)
{
    (void)in_sizes; (void)n_in; (void)out_size; (void)ws_size;
    const float* x      = (const float*)d_in[0];
    const float* il1_w1 = (const float*)d_in[1];
    const float* il1_b1 = (const float*)d_in[2];
    const float* il1_w2 = (const float*)d_in[3];
    const float* il1_b2 = (const float*)d_in[4];
    const float* fc_w   = (const float*)d_in[5];
    const float* fc_b   = (const float*)d_in[6];
    const float* ol1_w1 = (const float*)d_in[7];
    const float* ol1_b1 = (const float*)d_in[8];
    const float* ol1_w2 = (const float*)d_in[9];
    const float* ol1_b2 = (const float*)d_in[10];
    const float* il2_w1 = (const float*)d_in[11];
    const float* il2_b1 = (const float*)d_in[12];
    const float* il2_w2 = (const float*)d_in[13];
    const float* il2_b2 = (const float*)d_in[14];
    const float* ol2_w1 = (const float*)d_in[15];
    const float* ol2_b1 = (const float*)d_in[16];
    const float* ol2_w2 = (const float*)d_in[17];
    const float* ol2_b2 = (const float*)d_in[18];
    float* out = (float*)d_out;

    // Scratch carve-out (~101 MB total; all intermediates stay L2-resident: 192 MB L2)
    char* base = (char*)d_ws;
    size_t off = 0;
    auto carve = [&](size_t bytes) {
        char* p = base + off;
        off = (off + bytes + 255) & ~(size_t)255;
        return p;
    };
    int*   graph = (int*)  carve((size_t)MM * KNN * sizeof(int));
    float* bh    = (float*)carve((size_t)MM * CC * sizeof(float));       // h, then t3
    float* bt    = (float*)carve((size_t)MM * CC * sizeof(float));       // t1, t2, f2g
    float* bcat  = (float*)carve((size_t)MM * 2 * CC * sizeof(float));   // cat, then x1
    float* bbig  = (float*)carve((size_t)MM * HH * sizeof(float));       // f1, g1

    auto gemm = [&](const float* X, const float* W, const float* bsrc,
                    const float* resid, float* Y, int Nd, int Kd, int act) {
        const int blocks = (MM / 32) * (Nd / 32) / 4;   // 4 waves (32x32 tiles) per block
        gemm_bias_act<<<blocks, 128, 0, stream>>>(X, W, bsrc, resid, Y, Nd, Kd, act);
    };

    // 1) KNN graph from fp32 WMMA similarity
    sim_topk<<<BB * (NN / 16), 32, 0, stream>>>(x, graph);
    // 2) node MLP: h = mlp(x)
    gemm(x,  il1_w1, il1_b1, nullptr, bt, CC, CC, 1);      // t1 = gelu(x W1^T + b1)
    gemm(bt, il1_w2, il1_b2, nullptr, bh, CC, CC, 0);      // h  = t1 W2^T + b2
    // 3) gather-max aggregation + interleave
    agg_cat<<<(MM * CC) / 256, 256, 0, stream>>>(bh, graph, bcat);
    gemm(bcat, fc_w, fc_b, nullptr, bt, CC, 2 * CC, 1);    // t2 = gelu(fc(cat))
    gemm(bt, ol1_w1, ol1_b1, nullptr, bh, CC, CC, 1);      // t3 = gelu(t2 W^T + b)
    float* x1 = bcat;                                      // cat dead -> reuse
    gemm(bh, ol1_w2, ol1_b2, x, x1, CC, CC, 0);            // x1 = t3 W^T + b + x
    // 4) FFN
    gemm(x1,   il2_w1, il2_b1, nullptr, bbig, HH, CC, 1);  // f1  = gelu(x1 W^T+b)   [M x 768]
    gemm(bbig, il2_w2, il2_b2, nullptr, bt,   CC, HH, 1);  // f2g = gelu(f1 W^T+b)   (external gelu fused)
    gemm(bt,   ol2_w1, ol2_b1, nullptr, bbig, HH, CC, 1);  // g1  = gelu(f2g W^T+b)
    gemm(bbig, ol2_w2, ol2_b2, x1, out, CC, HH, 0);        // out = g1 W^T + b + x1
}